// AffinityLoss_90400471646752
// MI455X (gfx1250) — compile-verified
//
#include <hip/hip_runtime.h>

typedef float v2f __attribute__((ext_vector_type(2)));
typedef float v8f __attribute__((ext_vector_type(8)));
typedef int   v4i __attribute__((vector_size(4 * sizeof(int))));   // matches builtin's pointee

#define B_N   16384
#define D_N   512
#define K_N   256
#define BM    64
#define KC    16
#define LDSTR (KC + 4)        // 20 floats: 80B rows -> 16B aligned, conflict-free b64 frags
#define NCH   (D_N / KC)      // 32 D-chunks

#define AS1 __attribute__((address_space(1)))
#define AS3 __attribute__((address_space(3)))

#if __has_builtin(__builtin_amdgcn_global_load_async_to_lds_b128)
#define HAVE_ASYNC_LDS 1
#else
#define HAVE_ASYNC_LDS 0
#endif

#if __has_builtin(__builtin_amdgcn_s_wait_asynccnt)
#define WAIT_ASYNC(n) __builtin_amdgcn_s_wait_asynccnt(n)
#else
#define WAIT_ASYNC(n) asm volatile("s_wait_asynccnt " #n ::: "memory")
#endif

#if HAVE_ASYNC_LDS
__device__ __forceinline__ void async_cp16(const float* gp, float* lp) {
    // AS1 pointer value == 64-bit VA; AS3 pointer value == low 32 bits of the
    // LDS aperture address (the byte offset into LDS).
    __builtin_amdgcn_global_load_async_to_lds_b128(
        (AS1 v4i*)(size_t)gp,
        (AS3 v4i*)(unsigned int)(size_t)lp,
        0, 0);
}
#endif

// ---- wave-per-row sum of squares (for zz[B] and cc[K]) ----
__global__ __launch_bounds__(256) void row_sumsq_kernel(const float* __restrict__ x,
                                                        float* __restrict__ out,
                                                        int ncols) {
    const int lane = threadIdx.x & 31;
    const int wave = threadIdx.x >> 5;
    const int row  = blockIdx.x * 8 + wave;
    const float* rp = x + (size_t)row * ncols;
    float s = 0.0f;
    #pragma unroll
    for (int c = 0; c < D_N; c += 128) {
        float4 v = *(const float4*)(rp + c + lane * 4);
        s += v.x * v.x + v.y * v.y + v.z * v.z + v.w * v.w;
    }
    #pragma unroll
    for (int m = 16; m >= 1; m >>= 1) s += __shfl_xor(s, m, 32);
    if (lane == 0) out[row] = s;
}

// ---- fused GEMM (f32 WMMA) + t-student + row-normalize ----
__global__ __launch_bounds__(256) void affinity_kernel(const float* __restrict__ z,
                                                       const float* __restrict__ cl,
                                                       const float* __restrict__ zz,
                                                       const float* __restrict__ cc,
                                                       float* __restrict__ out) {
    __shared__ float Zs[2][BM * LDSTR];    //  2 x  5.0 KB
    __shared__ float Cs[2][K_N * LDSTR];   //  2 x 20.0 KB
    __shared__ float sumbuf[2][BM];
    __shared__ float invbuf[BM];

    const int tid    = threadIdx.x;
    const int lane   = tid & 31;
    const int wave   = tid >> 5;
    const int wave_m = wave & 3;      // 4 row groups of 16
    const int wave_n = wave >> 2;     // 2 col groups of 128
    const int laneN  = lane & 15;
    const int half   = lane >> 4;
    const int M0     = wave_m * 16;
    const int N0     = wave_n * 128;
    const int rowBase = blockIdx.x * BM;

    v8f acc[8];
    const v8f vzero = {0.f, 0.f, 0.f, 0.f, 0.f, 0.f, 0.f, 0.f};
    #pragma unroll
    for (int t = 0; t < 8; ++t) acc[t] = vzero;

    // staging geometry: one float4 per thread for Z, four for C, per chunk
    const int sr = tid >> 2;           // 0..63  (Z row / C row within group)
    const int sc = (tid & 3) * 4;      // float4 column within chunk
    const float* zg = z  + (size_t)(rowBase + sr) * D_N + sc;
    const float* cg = cl + (size_t)sr * D_N + sc;

    const int aRow = M0 + laneN;       // A frag: M = lane%16
    const int kOff = half * 2;         // lanes 16-31 hold K+2, K+3

#if HAVE_ASYNC_LDS
    // prologue: chunks 0 and 1 in flight
    #pragma unroll
    for (int pc = 0; pc < 2; ++pc) {
        const int kk = pc * KC;
        async_cp16(zg + kk, &Zs[pc][sr * LDSTR + sc]);
        #pragma unroll
        for (int i = 0; i < 4; ++i)
            async_cp16(cg + (size_t)(64 * i) * D_N + kk,
                       &Cs[pc][(64 * i + sr) * LDSTR + sc]);
    }
#endif

    for (int c = 0; c < NCH; ++c) {
        const int buf = c & 1;
#if HAVE_ASYNC_LDS
        if (c + 1 < NCH) { WAIT_ASYNC(5); } else { WAIT_ASYNC(0); }
        __syncthreads();
#else
        {
            const int kk = c * KC;
            float4 vz = *(const float4*)(zg + kk);
            *(float4*)(&Zs[buf][sr * LDSTR + sc]) = vz;
            #pragma unroll
            for (int i = 0; i < 4; ++i) {
                float4 vc = *(const float4*)(cg + (size_t)(64 * i) * D_N + kk);
                *(float4*)(&Cs[buf][(64 * i + sr) * LDSTR + sc]) = vc;
            }
        }
        __syncthreads();
#endif
        const float* Zb = &Zs[buf][0];
        const float* Cb = &Cs[buf][0];
        #pragma unroll
        for (int k4 = 0; k4 < KC; k4 += 4) {
            const v2f a = *(const v2f*)(&Zb[aRow * LDSTR + k4 + kOff]);
            #pragma unroll
            for (int t = 0; t < 8; ++t) {
                const int col = N0 + 16 * t + laneN;
                const v2f b = *(const v2f*)(&Cb[col * LDSTR + k4 + kOff]);
                acc[t] = __builtin_amdgcn_wmma_f32_16x16x4_f32(
                    false, a, false, b, (short)0, acc[t], false, false);
            }
        }
        __syncthreads();
#if HAVE_ASYNC_LDS
        if (c + 2 < NCH) {
            const int kk = (c + 2) * KC;
            async_cp16(zg + kk, &Zs[buf][sr * LDSTR + sc]);
            #pragma unroll
            for (int i = 0; i < 4; ++i)
                async_cp16(cg + (size_t)(64 * i) * D_N + kk,
                           &Cs[buf][(64 * i + sr) * LDSTR + sc]);
        }
#endif
    }

    // ---- epilogue: d = zz - 2*s + cc, q = 1/(1+max(d,0)), row-normalize ----
    float zzr[8], ccr[8];
    #pragma unroll
    for (int j = 0; j < 8; ++j) zzr[j] = zz[rowBase + M0 + j + 8 * half];
    #pragma unroll
    for (int t = 0; t < 8; ++t) ccr[t] = cc[N0 + 16 * t + laneN];

    #pragma unroll
    for (int t = 0; t < 8; ++t) {
        #pragma unroll
        for (int j = 0; j < 8; ++j) {
            float d = zzr[j] - 2.0f * acc[t][j] + ccr[t];
            d = fmaxf(d, 0.0f);
            acc[t][j] = 1.0f / (1.0f + d);   // ALPHA=1 -> exponent -1
        }
    }

    // per-row partial sums over this wave's 128 columns
    #pragma unroll
    for (int j = 0; j < 8; ++j) {
        float p = 0.0f;
        #pragma unroll
        for (int t = 0; t < 8; ++t) p += acc[t][j];
        #pragma unroll
        for (int m = 1; m <= 8; m <<= 1) p += __shfl_xor(p, m, 32);  // within 16-lane halves
        if (laneN == 0) sumbuf[wave_n][M0 + j + 8 * half] = p;
    }
    __syncthreads();
    if (tid < BM) invbuf[tid] = 1.0f / (sumbuf[0][tid] + sumbuf[1][tid]);
    __syncthreads();

    #pragma unroll
    for (int j = 0; j < 8; ++j) {
        const int rowL = M0 + j + 8 * half;
        const float inv = invbuf[rowL];
        const size_t rbase = (size_t)(rowBase + rowL) * K_N;
        #pragma unroll
        for (int t = 0; t < 8; ++t)
            out[rbase + N0 + 16 * t + laneN] = acc[t][j] * inv;
    }
}

extern "C" void kernel_launch(void* const* d_in, const int* in_sizes, int n_in,
                              void* d_out, int out_size, void* d_ws, size_t ws_size,
                              hipStream_t stream) {
    const float* z  = (const float*)d_in[0];   // z_inputs [B, D]
    // d_in[1] = x_inputs: unused by the reference
    const float* cl = (const float*)d_in[2];   // clusters [K, D]
    float* out = (float*)d_out;

    float* zzb = (float*)d_ws;                 // [B]
    float* ccb = zzb + B_N;                    // [K]

    row_sumsq_kernel<<<B_N / 8, 256, 0, stream>>>(z, zzb, D_N);
    row_sumsq_kernel<<<K_N / 8, 256, 0, stream>>>(cl, ccb, D_N);
    affinity_kernel<<<B_N / BM, 256, 0, stream>>>(z, cl, zzb, ccb, out);
}